// GAT_88476326298057
// MI455X (gfx1250) — compile-verified
//
#include <hip/hip_runtime.h>
#include <math.h>

// ---------------- problem constants (match reference) ----------------
#define NNODES   100000
#define NEDGES   1600000
#define NFEAT    128
#define NHID     128
#define NCLASS   64
#define NHEAD    4
#define DHEAD    (NHID / NHEAD)   // 32
#define NHIDLAY  2                // NLAYER-1 hidden GAT layers
#define LRELU_A  0.2f
#define KDIM     128              // all GEMMs have K = 128

typedef float v2f __attribute__((ext_vector_type(2)));
typedef float v8f __attribute__((ext_vector_type(8)));
typedef unsigned int u32x4 __attribute__((ext_vector_type(4)));
typedef int i32x8 __attribute__((ext_vector_type(8)));
typedef int i32x4 __attribute__((ext_vector_type(4)));

#define HAVE_TDM 0
#if defined(__has_builtin)
#if __has_builtin(__builtin_amdgcn_tensor_load_to_lds) && __has_builtin(__builtin_amdgcn_s_wait_tensorcnt)
#undef HAVE_TDM
#define HAVE_TDM 1
#endif
#endif

// ---------------- small helpers ----------------
__device__ __forceinline__ float elu_f(float v) {
    return v > 0.0f ? v : (__expf(v) - 1.0f);
}
__device__ __forceinline__ float lrelu_f(float v) {
    return v > 0.0f ? v : LRELU_A * v;
}
// order-preserving float<->uint mapping for atomicMax on signed floats
__device__ __forceinline__ unsigned fenc(float f) {
    unsigned u = __float_as_uint(f);
    return (u & 0x80000000u) ? ~u : (u | 0x80000000u);
}
__device__ __forceinline__ float fdec(unsigned u) {
    return __uint_as_float((u & 0x80000000u) ? (u ^ 0x80000000u) : ~u);
}
#define NEG_INF_ENC 0x007FFFFFu   // fenc(-inf)

#if HAVE_TDM
// Issue a TDM contiguous copy of `totalElems` f32 (totalElems <= 65535) from
// global memory to LDS. 2D descriptor: dim0 = tile_dim0 = totalElems, dim1 = 1.
__device__ __forceinline__ void tdm_load_1d(unsigned lds_off, const void* gptr,
                                            unsigned totalElems) {
    unsigned long long ga = (unsigned long long)(uintptr_t)gptr;
    u32x4 g0;
    g0[0] = 1u;                                          // count=1, user mode
    g0[1] = lds_off;                                     // LDS byte address
    g0[2] = (unsigned)ga;                                // global_addr[31:0]
    g0[3] = ((unsigned)(ga >> 32) & 0x01FFFFFFu)         // global_addr[56:32]
            | 0x80000000u;                               // type=2 ("image")
    i32x8 g1;
    g1[0] = 0x00020000;                                  // data_size=4B, mask=0
    g1[1] = (int)((totalElems & 0xFFFFu) << 16);         // tensor_dim0[15:0]
    g1[2] = (int)(((totalElems >> 16) & 0xFFFFu)         // tensor_dim0[31:16]
            | (1u << 16));                               // tensor_dim1[15:0]=1
    g1[3] = (int)((totalElems & 0xFFFFu) << 16);         // tile_dim0 (16-bit)
    g1[4] = 1;                                           // tile_dim1=1, tile_dim2=0
    g1[5] = (int)totalElems;                             // dim0_stride[31:0]
    g1[6] = 0;
    g1[7] = 0;
    i32x4 z4 = {0, 0, 0, 0};
    i32x8 z8 = {0, 0, 0, 0, 0, 0, 0, 0};
    // 6-arg form (this toolchain): (g0, g1, g2, g3, g4, cpol)
    __builtin_amdgcn_tensor_load_to_lds(g0, g1, z4, z4, z8, 0);
}
#endif

// ---------------- fill kernels ----------------
__global__ void fill_f32(float* __restrict__ p, float v, long long n) {
    long long i = (long long)blockIdx.x * blockDim.x + threadIdx.x;
    if (i < n) p[i] = v;
}
__global__ void fill_u32(unsigned* __restrict__ p, unsigned v, long long n) {
    long long i = (long long)blockIdx.x * blockDim.x + threadIdx.x;
    if (i < n) p[i] = v;
}

// ---------------- WMMA GEMM with LDS-staged B (via TDM) ----------------------
// C[M, Nc] = act(A[M,K] @ B + bias), Nc = NTILES*16, K = KDIM = 128.
// MODE 0: dense B[K][Nc].  MODE 1: per-head B[H][K][32], Nc = H*32.
// Block = 256 threads = 8 waves; wave w computes the full 16 x Nc row strip of
// m-tile (blockIdx.x*8 + w). B is DMA'd to LDS once per block and read via DS.
template <int NTILES, int MODE>
__global__ __launch_bounds__(256)
void gemm_wmma_smem(const float* __restrict__ A,
                    const float* __restrict__ B,
                    const float* __restrict__ bias,
                    float* __restrict__ C,
                    int M, int useBias, int act) {
    constexpr int NC = NTILES * 16;
    constexpr int BS_ELEMS = KDIM * NC;   // dense: K*Nc; heads: 4*128*32 == K*Nc
    __shared__ float Bs[BS_ELEMS];

    // ---- stage B into LDS (all threads reach the barrier) ----
#if HAVE_TDM
    if ((threadIdx.x >> 5) == 0) {
        tdm_load_1d((unsigned)(uintptr_t)Bs, B, BS_ELEMS);
        __builtin_amdgcn_s_wait_tensorcnt(0);
    }
#else
    for (int i = threadIdx.x; i < BS_ELEMS; i += blockDim.x) Bs[i] = B[i];
#endif
    __syncthreads();

    const int wave = threadIdx.x >> 5;
    const int lane = threadIdx.x & 31;
    const int mt   = blockIdx.x * 8 + wave;
    const int Mtiles = M >> 4;
    if (mt >= Mtiles) return;             // wave-uniform exit, EXEC stays all-1s

    const int m0   = mt << 4;
    const int half = lane >> 4;           // selects K sub-pair {0,1} vs {2,3}
    const int l    = lane & 15;

    const float* __restrict__ Arow = A + (size_t)(m0 + l) * KDIM;
    v8f acc[NTILES];
#pragma unroll
    for (int nt = 0; nt < NTILES; ++nt) acc[nt] = (v8f){};

#pragma unroll 4
    for (int k0 = 0; k0 < KDIM; k0 += 4) {
        const int ka = k0 + 2 * half;
        __builtin_prefetch(Arow + k0 + 64, 0, 1);
        const float2 av = *(const float2*)(Arow + ka);
        v2f a;
        a.x = av.x;
        a.y = av.y;
#pragma unroll
        for (int nt = 0; nt < NTILES; ++nt) {
            const int n = nt * 16 + l;
            int bi0, bi1;
            if (MODE == 0) {
                bi0 = ka * NC + n;
                bi1 = (ka + 1) * NC + n;
            } else {  // per-head block-diagonal: head = n>>5, Dh = 32
                const int head = n >> 5;
                const int bc   = n & 31;
                bi0 = (head * KDIM + ka) * 32 + bc;
                bi1 = (head * KDIM + ka + 1) * 32 + bc;
            }
            v2f b;
            b.x = Bs[bi0];
            b.y = Bs[bi1];
            acc[nt] = __builtin_amdgcn_wmma_f32_16x16x4_f32(
                false, a, false, b, (short)0, acc[nt], false, false);
        }
    }

#pragma unroll
    for (int nt = 0; nt < NTILES; ++nt) {
#pragma unroll
        for (int j = 0; j < 8; ++j) {
            const int r  = m0 + half * 8 + j;
            const int cc = nt * 16 + l;
            float v = acc[nt][j];
            if (useBias) v += bias[cc];
            if (act)     v  = elu_f(v);
            C[(size_t)r * NC + cc] = v;
        }
    }
}

// ---------------- attention score vectors:  s_src/s_dst [n, H] ----------------
// a layout: [H][2][D]
__global__ void scores_kernel(const float* __restrict__ h,
                              const float* __restrict__ a,
                              float* __restrict__ s_src,
                              float* __restrict__ s_dst,
                              int n, int H, int D) {
    int i = blockIdx.x * blockDim.x + threadIdx.x;
    if (i >= n) return;
    const float* __restrict__ hv = h + (size_t)i * D;
    for (int hd = 0; hd < H; ++hd) {
        const float* __restrict__ as = a + ((size_t)hd * 2) * D;
        const float* __restrict__ ad = as + D;
        float ss = 0.0f, sd = 0.0f;
        for (int k = 0; k < D; k += 4) {
            const float4 v = *(const float4*)(hv + k);
            const float4 s = *(const float4*)(as + k);
            const float4 d = *(const float4*)(ad + k);
            ss = fmaf(v.x, s.x, fmaf(v.y, s.y, fmaf(v.z, s.z, fmaf(v.w, s.w, ss))));
            sd = fmaf(v.x, d.x, fmaf(v.y, d.y, fmaf(v.z, d.z, fmaf(v.w, d.w, sd))));
        }
        s_src[(size_t)i * H + hd] = ss;
        s_dst[(size_t)i * H + hd] = sd;
    }
}

// ---------------- edge pass 1: segment max of leaky-relu logits ----------------
__global__ void edge_max_kernel(const int* __restrict__ row,
                                const int* __restrict__ col,
                                const float* __restrict__ ss,
                                const float* __restrict__ sd,
                                unsigned* __restrict__ m_u,
                                int ne, int H) {
    int e = blockIdx.x * blockDim.x + threadIdx.x;
    if (e >= ne) return;
    const int r = row[e], c = col[e];
    for (int hd = 0; hd < H; ++hd) {
        float lg = lrelu_f(ss[(size_t)r * H + hd] + sd[(size_t)c * H + hd]);
        atomicMax(&m_u[(size_t)r * H + hd], fenc(lg));
    }
}

// ---------------- edge pass 2: exp + segment sum, stash exp per edge ----------
__global__ void edge_expsum_kernel(const int* __restrict__ row,
                                   const int* __restrict__ col,
                                   const float* __restrict__ ss,
                                   const float* __restrict__ sd,
                                   const unsigned* __restrict__ m_u,
                                   float* __restrict__ ealpha,
                                   float* __restrict__ ssum,
                                   int ne, int H) {
    int e = blockIdx.x * blockDim.x + threadIdx.x;
    if (e >= ne) return;
    const int r = row[e], c = col[e];
    for (int hd = 0; hd < H; ++hd) {
        float lg = lrelu_f(ss[(size_t)r * H + hd] + sd[(size_t)c * H + hd]);
        float ex = __expf(lg - fdec(m_u[(size_t)r * H + hd]));
        ealpha[(size_t)e * H + hd] = ex;
        atomicAdd(&ssum[(size_t)r * H + hd], ex);
    }
}

// ---------------- edge pass 3: normalized weighted scatter-aggregate ----------
// out[row, t] += (e/sum) * g[col, t]; thread handles 4 consecutive columns.
__global__ void edge_agg_kernel(const int* __restrict__ row,
                                const int* __restrict__ col,
                                const float* __restrict__ ealpha,
                                const float* __restrict__ ssum,
                                const float* __restrict__ g,
                                float* __restrict__ outp,
                                int ne, int H, int Dh) {
    const int Dtot = H * Dh;
    const int QT = Dtot >> 2;             // float4 chunks per edge
    long long idx = (long long)blockIdx.x * blockDim.x + threadIdx.x;
    if (idx >= (long long)ne * QT) return;
    const int e = (int)(idx / QT);
    const int q = (int)(idx - (long long)e * QT);
    const int t = q << 2;
    const int hd = t / Dh;
    const int r = row[e], c = col[e];
    const float alpha = ealpha[(size_t)e * H + hd] / ssum[(size_t)r * H + hd];
    const float4 gv = *(const float4*)(g + (size_t)c * Dtot + t);
    float* op = outp + (size_t)r * Dtot + t;
    atomicAdd(op + 0, alpha * gv.x);
    atomicAdd(op + 1, alpha * gv.y);
    atomicAdd(op + 2, alpha * gv.z);
    atomicAdd(op + 3, alpha * gv.w);
}

// ---------------- elementwise ELU ----------------
__global__ void elu_kernel(float* __restrict__ p, long long n) {
    long long i = (long long)blockIdx.x * blockDim.x + threadIdx.x;
    if (i < n) p[i] = elu_f(p[i]);
}

// ---------------- host orchestration ----------------
static inline int cdiv(long long a, long long b) { return (int)((a + b - 1) / b); }

extern "C" void kernel_launch(void* const* d_in, const int* in_sizes, int n_in,
                              void* d_out, int out_size, void* d_ws, size_t ws_size,
                              hipStream_t stream) {
    (void)in_sizes; (void)n_in; (void)out_size; (void)ws_size;
    const float* x     = (const float*)d_in[0];
    const int*   ei    = (const int*)  d_in[1];
    const float* Win   = (const float*)d_in[2];
    const float* b_in  = (const float*)d_in[3];
    const float* a_hid = (const float*)d_in[4];   // [2,4,2,128]
    const float* W_hid = (const float*)d_in[5];   // [2,4,128,32]
    const float* a_out = (const float*)d_in[6];   // [2,128]
    const float* W_out = (const float*)d_in[7];   // [128,64]
    float* out = (float*)d_out;

    const int* row = ei;
    const int* col = ei + NEDGES;

    // workspace layout (floats)
    float* ws = (float*)d_ws;
    const size_t SZ_H = (size_t)NNODES * NHID;
    float*    h      = ws;                         // [N,128]
    float*    hnew   = h      + SZ_H;              // [N,128]
    float*    g      = hnew   + SZ_H;              // [N,128] (>= N*64)
    float*    s_src  = g      + SZ_H;              // [N,4]
    float*    s_dst  = s_src  + (size_t)NNODES * NHEAD;
    unsigned* m_u    = (unsigned*)(s_dst + (size_t)NNODES * NHEAD); // [N,4]
    float*    ssum   = (float*)(m_u + (size_t)NNODES * NHEAD);      // [N,4]
    float*    ealpha = ssum + (size_t)NNODES * NHEAD;               // [E,4]

    const int TB = 256;
    const int Mtiles = NNODES / 16;                // 6250
    const int gemmBlocks = cdiv(Mtiles, 8);        // 8 waves (m-tiles) per block

    // ---- layer 0: h = elu(x @ Win + b_in) ----
    gemm_wmma_smem<8, 0><<<gemmBlocks, 256, 0, stream>>>(
        x, Win, b_in, h, NNODES, /*useBias=*/1, /*act=elu*/1);

    // ---- hidden GAT layers ----
    for (int l = 0; l < NHIDLAY; ++l) {
        const float* a_l = a_hid + (size_t)l * NHEAD * 2 * NHID;
        const float* W_l = W_hid + (size_t)l * NHEAD * NHID * DHEAD;

        scores_kernel<<<cdiv(NNODES, TB), TB, 0, stream>>>(
            h, a_l, s_src, s_dst, NNODES, NHEAD, NHID);

        gemm_wmma_smem<8, 1><<<gemmBlocks, 256, 0, stream>>>(
            h, W_l, nullptr, g, NNODES, /*useBias=*/0, /*act=*/0);

        long long nmh = (long long)NNODES * NHEAD;
        fill_u32<<<cdiv(nmh, TB), TB, 0, stream>>>(m_u, NEG_INF_ENC, nmh);
        fill_f32<<<cdiv(nmh, TB), TB, 0, stream>>>(ssum, 0.0f, nmh);
        fill_f32<<<cdiv((long long)SZ_H, TB), TB, 0, stream>>>(hnew, 0.0f, (long long)SZ_H);

        edge_max_kernel<<<cdiv(NEDGES, TB), TB, 0, stream>>>(
            row, col, s_src, s_dst, m_u, NEDGES, NHEAD);
        edge_expsum_kernel<<<cdiv(NEDGES, TB), TB, 0, stream>>>(
            row, col, s_src, s_dst, m_u, ealpha, ssum, NEDGES, NHEAD);

        long long work = (long long)NEDGES * (NHID / 4);
        edge_agg_kernel<<<cdiv(work, TB), TB, 0, stream>>>(
            row, col, ealpha, ssum, g, hnew, NEDGES, NHEAD, DHEAD);

        elu_kernel<<<cdiv((long long)SZ_H, TB), TB, 0, stream>>>(hnew, (long long)SZ_H);

        float* tmp = h; h = hnew; hnew = tmp;      // ping-pong
    }

    // ---- final layer: single head -> NCLASS, no activation ----
    scores_kernel<<<cdiv(NNODES, TB), TB, 0, stream>>>(
        h, a_out, s_src, s_dst, NNODES, /*H=*/1, NHID);

    gemm_wmma_smem<4, 0><<<gemmBlocks, 256, 0, stream>>>(
        h, W_out, nullptr, g, NNODES, /*useBias=*/0, /*act=*/0);

    long long nm1 = (long long)NNODES;
    fill_u32<<<cdiv(nm1, TB), TB, 0, stream>>>(m_u, NEG_INF_ENC, nm1);
    fill_f32<<<cdiv(nm1, TB), TB, 0, stream>>>(ssum, 0.0f, nm1);
    long long osz = (long long)NNODES * NCLASS;
    fill_f32<<<cdiv(osz, TB), TB, 0, stream>>>(out, 0.0f, osz);

    edge_max_kernel<<<cdiv(NEDGES, TB), TB, 0, stream>>>(
        row, col, s_src, s_dst, m_u, NEDGES, 1);
    edge_expsum_kernel<<<cdiv(NEDGES, TB), TB, 0, stream>>>(
        row, col, s_src, s_dst, m_u, ealpha, ssum, NEDGES, 1);

    long long work = (long long)NEDGES * (NCLASS / 4);
    edge_agg_kernel<<<cdiv(work, TB), TB, 0, stream>>>(
        row, col, ealpha, ssum, g, out, NEDGES, /*H=*/1, /*Dh=*/NCLASS);
}